// SpectralConv2d_59390807769306
// MI455X (gfx1250) — compile-verified
//
#include <hip/hip_runtime.h>
#include <math.h>

// SpectralConv2d (FNO layer): B=16, Ci=Co=64, H=W=256, modes 16x16, scale 1/4096.
// Truncated DFTs expressed as f32 WMMA GEMMs (only 2*16*16 modes are live, so
// no full FFT is needed).  ky set S = {0..15} U {240..255} (32 values).

typedef __attribute__((ext_vector_type(2))) float v2f;
typedef __attribute__((ext_vector_type(8))) float v8f;
typedef __attribute__((ext_vector_type(4))) unsigned int v4u;
typedef __attribute__((ext_vector_type(8))) unsigned int v8u;

#define PI_F 3.14159265358979323846f
#define SCALE_W (1.0f / 4096.0f)     // 1/(in_ch*out_ch)
#define INV_HW (1.0f / 65536.0f)     // irfft2 normalization

__device__ __forceinline__ v8f wmma_f32(v2f a, v2f b, v8f c) {
  // D = A(16x4 f32) * B(4x16 f32) + C(16x16 f32)
  return __builtin_amdgcn_wmma_f32_16x16x4_f32(
      /*neg_a=*/false, a, /*neg_b=*/false, b,
      /*c_mod=*/(short)0, c, /*reuse_a=*/false, /*reuse_b=*/false);
}

__device__ __forceinline__ int ky_of(int r) {  // r in 0..31 -> ky
  return (r < 16) ? r : (224 + r);             // 16..31 -> 240..255
}

__device__ __forceinline__ unsigned lds_off(const void* p) {
  return (unsigned)(unsigned long long)p;  // low 32 bits of flat LDS address
}

// Tensor Data Mover: DMA a (rows x width) f32 tile with row stride
// `strideElems` from global memory into LDS, packed row-major.  Issued once
// per wave (EXEC ignored); tracked on TENSORcnt.  D# layout per CDNA5 ISA
// ch.8 (group0: count/lds_addr/global_addr/type, group1: dims/strides).
__device__ __forceinline__ void tdm_load_f32(unsigned ldsOff, const void* gptr,
                                             unsigned width, unsigned rows,
                                             unsigned long long strideElems) {
  const unsigned long long ga = (unsigned long long)gptr;
  v4u g0;
  g0[0] = 1u;                                         // count=1, user descriptor
  g0[1] = ldsOff;                                     // lds_addr (bytes)
  g0[2] = (unsigned)ga;                               // global_addr[31:0]
  g0[3] = (unsigned)((ga >> 32) & 0x1ffffffu) | (2u << 30);  // addr[56:32]|type=2
  v8u g1;
  g1[0] = 2u << 16;                                   // data_size = 4 bytes
  g1[1] = (width & 0xffffu) << 16;                    // tensor_dim0[15:0]
  g1[2] = (width >> 16) | ((rows & 0xffffu) << 16);   // dim0 hi | tensor_dim1 lo
  g1[3] = (rows >> 16) | ((width & 0xffffu) << 16);   // dim1 hi | tile_dim0
  g1[4] = rows & 0xffffu;                             // tile_dim1 (tile_dim2=0)
  g1[5] = (unsigned)strideElems;                      // tensor_dim0_stride lo
  g1[6] = (unsigned)(strideElems >> 32) & 0xffffu;    // stride hi (dim1_stride=0)
  g1[7] = 0u;
  asm volatile("tensor_load_to_lds %0, %1" :: "s"(g0), "s"(g1) : "memory");
}

// ---------------------------------------------------------------------------
// Kernel 1: forward truncated DFT.  One block per (b,ci) image (1024 blocks).
//   Phase A (WMMA): D[64x256] = T[64x256] @ x[256x256]
//       T rows 0..31  = cos(2*pi*ky*y/256)   (real part rows)
//       T rows 32..63 = -sin(2*pi*ky*y/256)  (imag part rows)
//   Phase B (VALU):  XF[kyi][kx] = sum_x D_complex[kyi][x] * e^{-2pi i kx x/256}
// ---------------------------------------------------------------------------
__global__ __launch_bounds__(256)
void kfwd(const float* __restrict__ x,
          float* __restrict__ XFre, float* __restrict__ XFim) {
  constexpr int LDT = 260;                 // padded row stride (bank-conflict free)
  __shared__ float Tt[64 * LDT];
  __shared__ float Ds[64 * LDT];

  const int img = blockIdx.x;              // b*64 + ci
  const float* __restrict__ xi = x + (size_t)img * 65536;
  const int t = threadIdx.x;
  const int lane = t & 31;
  const int wv = t >> 5;                   // 0..7
  const int hi = lane >> 4;                // 0/1
  const int li = lane & 15;

  for (int i = t; i < 64 * 256; i += 256) {
    int r = i >> 8, y = i & 255;
    int ky = ky_of(r & 31);
    float th = (2.0f * PI_F / 256.0f) * (float)((ky * y) & 255);
    float s, c;
    __sincosf(th, &s, &c);
    Tt[r * LDT + y] = (r < 32) ? c : -s;
  }
  __syncthreads();

  // Phase A: wave wv owns x-columns [wv*32, wv*32+31] = two N-tiles; each
  // A-fragment feeds both N-tiles (halves LDS traffic per wmma).
  const int col0 = wv * 32 + li;
  const int col1 = col0 + 16;
  v8f a00 = {}, a01 = {}, a02 = {}, a03 = {};
  v8f a10 = {}, a11 = {}, a12 = {}, a13 = {};
  for (int k = 0; k < 256; k += 4) {
    const int kr = k + 2 * hi;
    v2f b0, b1;
    b0.x = xi[(size_t)kr * 256 + col0];
    b0.y = xi[(size_t)(kr + 1) * 256 + col0];
    b1.x = xi[(size_t)kr * 256 + col1];
    b1.y = xi[(size_t)(kr + 1) * 256 + col1];
    __builtin_prefetch(&xi[(size_t)(kr + 8) * 256 + col0], 0, 1);
    v2f t0 = *(const v2f*)&Tt[(0  + li) * LDT + kr];
    v2f t1 = *(const v2f*)&Tt[(16 + li) * LDT + kr];
    v2f t2 = *(const v2f*)&Tt[(32 + li) * LDT + kr];
    v2f t3 = *(const v2f*)&Tt[(48 + li) * LDT + kr];
    a00 = wmma_f32(t0, b0, a00);
    a10 = wmma_f32(t0, b1, a10);
    a01 = wmma_f32(t1, b0, a01);
    a11 = wmma_f32(t1, b1, a11);
    a02 = wmma_f32(t2, b0, a02);
    a12 = wmma_f32(t2, b1, a12);
    a03 = wmma_f32(t3, b0, a03);
    a13 = wmma_f32(t3, b1, a13);
  }
#pragma unroll
  for (int r = 0; r < 8; ++r) {
    const int m = r + 8 * hi;
    Ds[(0  + m) * LDT + col0] = a00[r];
    Ds[(0  + m) * LDT + col1] = a10[r];
    Ds[(16 + m) * LDT + col0] = a01[r];
    Ds[(16 + m) * LDT + col1] = a11[r];
    Ds[(32 + m) * LDT + col0] = a02[r];
    Ds[(32 + m) * LDT + col1] = a12[r];
    Ds[(48 + m) * LDT + col0] = a03[r];
    Ds[(48 + m) * LDT + col1] = a13[r];
  }
  __syncthreads();

  // Phase B: project onto kx modes 0..15.  X1 = a + i b, times e^{-i th}:
  //   re += a*c + b*s ; im += b*c - a*s
  for (int o = t; o < 512; o += 256) {
    const int kyi = o >> 4, kx = o & 15;
    float ca, sa;
    __sincosf((2.0f * PI_F / 256.0f) * (float)kx, &sa, &ca);
    float sre = 0.0f, sim = 0.0f, c = 1.0f, s = 0.0f;
    for (int xx = 0; xx < 256; ++xx) {
      if ((xx & 63) == 0) {  // resync rotation to limit fp32 drift
        float th = (2.0f * PI_F / 256.0f) * (float)((kx * xx) & 255);
        __sincosf(th, &s, &c);
      }
      const float a = Ds[kyi * LDT + xx];
      const float b = Ds[(kyi + 32) * LDT + xx];
      sre += a * c + b * s;
      sim += b * c - a * s;
      const float cn = c * ca - s * sa;
      s = s * ca + c * sa;
      c = cn;
    }
    const size_t idx = ((size_t)img * 32 + kyi) * 16 + kx;
    XFre[idx] = sre;
    XFim[idx] = sim;
  }
}

// ---------------------------------------------------------------------------
// Kernel 2: per-mode complex channel mix.  512 blocks = 2 blocks * 16 * 16.
//   Y[b,co] = SCALE * sum_ci X[b,ci] * W[ci,co]   (complex)
// X gather across (b,ci) is a strided 1024x1 tile -> Tensor Data Mover.
// ---------------------------------------------------------------------------
__global__ __launch_bounds__(256)
void kmix(const float* __restrict__ r1, const float* __restrict__ i1,
          const float* __restrict__ r2, const float* __restrict__ i2,
          const float* __restrict__ XFre, const float* __restrict__ XFim,
          float* __restrict__ YFre, float* __restrict__ YFim) {
  __shared__ float Xr[16 * 64], Xi[16 * 64];
  __shared__ float Wr[64 * 64], Wi[64 * 64];

  const int m = blockIdx.x;                 // 0..511
  const int blk = m >> 8;                   // 0: ky 0..15, 1: ky 240..255
  const int my = (m >> 4) & 15;
  const int mx = m & 15;
  const int kyi = blk * 16 + my;
  const int t = threadIdx.x;
  const float* __restrict__ wr = blk ? r2 : r1;
  const float* __restrict__ wi = blk ? i2 : i1;

  if (t < 32) {  // wave 0: TDM gathers X[b,ci] (1024 rows, stride 512 floats)
    tdm_load_f32(lds_off(Xr), XFre + kyi * 16 + mx, 1u, 1024u, 512ull);
    tdm_load_f32(lds_off(Xi), XFim + kyi * 16 + mx, 1u, 1024u, 512ull);
  }
  for (int i = t; i < 4096; i += 256) {     // i = ci*64 + co
    const size_t idx = (size_t)i * 256 + my * 16 + mx;
    Wr[i] = wr[idx];
    Wi[i] = wi[idx];
  }
  if (t < 32) __builtin_amdgcn_s_wait_tensorcnt(0);
  __syncthreads();

  for (int o = t; o < 1024; o += 256) {     // o = b*64 + co
    const int b = o >> 6, co = o & 63;
    float yr = 0.0f, yi = 0.0f;
    for (int ci = 0; ci < 64; ++ci) {
      const float xr = Xr[b * 64 + ci], xim = Xi[b * 64 + ci];
      const float wrv = Wr[ci * 64 + co], wiv = Wi[ci * 64 + co];
      yr += xr * wrv - xim * wiv;
      yi += xr * wiv + xim * wrv;
    }
    const size_t idx = ((size_t)o * 32 + kyi) * 16 + mx;
    YFre[idx] = yr * SCALE_W;
    YFim[idx] = yi * SCALE_W;
  }
}

// ---------------------------------------------------------------------------
// Kernel 3: inverse.  One block per (b,co) image (1024 blocks).
//   Expand: G[kyi][x] = sum_kx w_kx * Y[kyi,kx] * e^{+2pi i kx x/256},
//           w_0 = 1, w_kx = 2 (Hermitian half-spectrum of irfft).
//   GEMM:   out[256x256] = Ty[256x64] @ Gcat[64x256]; Re taken by construction:
//           Ty cols 0..31 = cos * INV_HW, cols 32..63 = -sin * INV_HW.
// ---------------------------------------------------------------------------
__global__ __launch_bounds__(256)
void kinv(const float* __restrict__ YFre, const float* __restrict__ YFim,
          float* __restrict__ out) {
  constexpr int LTY = 68;                   // padded stride for Ty (256x64)
  constexpr int LG = 272;                   // padded stride for G (64x256)
  __shared__ float Ty[256 * LTY];
  __shared__ float G[64 * LG];
  __shared__ float Ys[2 * 512];             // [re(512) | im(512)], idx = kyi*16+kx

  const int img = blockIdx.x;               // b*64 + co
  const int t = threadIdx.x;
  const int lane = t & 31;
  const int wv = t >> 5;
  const int hi = lane >> 4;
  const int li = lane & 15;

  if (t < 32) {  // wave 0: TDM fetch of this image's 512 mode coefficients
    tdm_load_f32(lds_off(Ys), YFre + (size_t)img * 512, 512u, 1u, 512ull);
    tdm_load_f32(lds_off(Ys + 512), YFim + (size_t)img * 512, 512u, 1u, 512ull);
  }
  {  // Ty: thread t fills row y = t (64 entries)
    const int y = t;
    for (int r = 0; r < 64; ++r) {
      const int ky = ky_of(r & 31);
      float th = (2.0f * PI_F / 256.0f) * (float)((ky * y) & 255);
      float s, c;
      __sincosf(th, &s, &c);
      Ty[y * LTY + r] = ((r < 32) ? c : -s) * INV_HW;
    }
  }
  if (t < 32) __builtin_amdgcn_s_wait_tensorcnt(0);
  __syncthreads();

  {  // G: thread t handles column x = t.  (a+ib)*e^{+i th}.
    const int xx = t;
    float ca, sa;
    __sincosf((2.0f * PI_F / 256.0f) * (float)xx, &sa, &ca);
    for (int kyi = 0; kyi < 32; ++kyi) {
      float gre = Ys[kyi * 16];             // kx = 0, weight 1
      float gim = Ys[512 + kyi * 16];
      float c = 1.0f, s = 0.0f;
      for (int kx = 1; kx < 16; ++kx) {
        const float cn = c * ca - s * sa;   // rotate to e^{i kx th0}
        s = s * ca + c * sa;
        c = cn;
        const float ar = Ys[kyi * 16 + kx];
        const float ai = Ys[512 + kyi * 16 + kx];
        gre += 2.0f * (ar * c - ai * s);
        gim += 2.0f * (ar * s + ai * c);
      }
      G[kyi * LG + xx] = gre;
      G[(32 + kyi) * LG + xx] = gim;
    }
  }
  __syncthreads();

  // GEMM: wave wv owns output rows [wv*32, wv*32+31]; each A-fragment is
  // reused across an N-tile pair.
  float* __restrict__ outi = out + (size_t)img * 65536;
  for (int mm = 0; mm < 2; ++mm) {
    const int m0 = (wv * 2 + mm) * 16;
    for (int np = 0; np < 8; ++np) {
      const int n0 = np * 32;
      v8f acc0 = {}, acc1 = {};
      for (int k = 0; k < 64; k += 4) {
        const int kr = k + 2 * hi;
        v2f af = *(const v2f*)&Ty[(m0 + li) * LTY + kr];
        v2f b0, b1;
        b0.x = G[kr * LG + n0 + li];
        b0.y = G[(kr + 1) * LG + n0 + li];
        b1.x = G[kr * LG + n0 + 16 + li];
        b1.y = G[(kr + 1) * LG + n0 + 16 + li];
        acc0 = wmma_f32(af, b0, acc0);
        acc1 = wmma_f32(af, b1, acc1);
      }
#pragma unroll
      for (int r = 0; r < 8; ++r) {
        const int row = m0 + r + 8 * hi;
        outi[(size_t)row * 256 + n0 + li] = acc0[r];
        outi[(size_t)row * 256 + n0 + 16 + li] = acc1[r];
      }
    }
  }
}

// ---------------------------------------------------------------------------
extern "C" void kernel_launch(void* const* d_in, const int* in_sizes, int n_in,
                              void* d_out, int out_size, void* d_ws, size_t ws_size,
                              hipStream_t stream) {
  (void)in_sizes; (void)n_in; (void)out_size; (void)ws_size;
  const float* x  = (const float*)d_in[0];
  const float* r1 = (const float*)d_in[1];
  const float* i1 = (const float*)d_in[2];
  const float* r2 = (const float*)d_in[3];
  const float* i2 = (const float*)d_in[4];

  float* ws = (float*)d_ws;
  const size_t NXF = (size_t)16 * 64 * 32 * 16;  // 524288 floats per plane
  float* XFre = ws;
  float* XFim = ws + NXF;
  float* YFre = ws + 2 * NXF;
  float* YFim = ws + 3 * NXF;

  kfwd<<<dim3(1024), dim3(256), 0, stream>>>(x, XFre, XFim);
  kmix<<<dim3(512), dim3(256), 0, stream>>>(r1, i1, r2, i2, XFre, XFim, YFre, YFim);
  kinv<<<dim3(1024), dim3(256), 0, stream>>>(YFre, YFim, (float*)d_out);
}